// BSModel_42210938585849
// MI455X (gfx1250) — compile-verified
//
#include <hip/hip_runtime.h>
#include <math.h>

// Elementwise Black-Scholes: memory-bound target (32 B/option, 268 MB total,
// ~11.5 us floor @ 23.3 TB/s). Round-1 version was compute-bound (~320
// instrs/option from precise ocml erf/log/exp/div). This version hand-rolls
// branchless math on raw CDNA5 trans ops (v_log_f32, v_exp_f32, v_rcp_f32,
// v_sqrt_f32) to ~60 VALU/option, putting ALU back under the HBM roof.
// Memory path: global_load_b128 th:TH_LOAD_HT (inputs stay L2-resident
// across graph replays; 134 MB < 192 MB L2) + non-temporal b128 stores
// (output stream bypasses L2, doesn't evict inputs).

typedef float v4f __attribute__((ext_vector_type(4)));

#define BS_R 0.02f
#define BS_Q 0.01f
#define LN2F 0.6931471805599453f
#define LOG2EF 1.4426950408889634f

// 4x 128-bit loads with high-priority-temporal hint, single loadcnt wait.
// Earlyclobber outputs: returned data may write result VGPRs before later
// loads in the clause read their address VGPRs.
__device__ __forceinline__ void load4_b128_ht(const v4f* a, const v4f* b,
                                              const v4f* c, const v4f* d,
                                              v4f& ra, v4f& rb, v4f& rc, v4f& rd) {
    asm volatile(
        "global_load_b128 %0, %4, off th:TH_LOAD_HT\n\t"
        "global_load_b128 %1, %5, off th:TH_LOAD_HT\n\t"
        "global_load_b128 %2, %6, off th:TH_LOAD_HT\n\t"
        "global_load_b128 %3, %7, off th:TH_LOAD_HT\n\t"
        "s_wait_loadcnt 0x0"
        : "=&v"(ra), "=&v"(rb), "=&v"(rc), "=&v"(rd)
        : "v"(a), "v"(b), "v"(c), "v"(d)
        : "memory");
}

// Branchless erf, Abramowitz-Stegun 7.1.26 (max abs err ~1.5e-7).
// ~13 VALU: |x|, rcp, 5 fma, 3 mul, v_exp_f32, fma, copysign.
__device__ __forceinline__ float erf_fast(float x) {
    float ax = __builtin_fabsf(x);
    float t  = __builtin_amdgcn_rcpf(__builtin_fmaf(0.3275911f, ax, 1.0f));
    float p  = __builtin_fmaf(1.061405429f, t, -1.453152027f);
    p = __builtin_fmaf(p, t, 1.421413741f);
    p = __builtin_fmaf(p, t, -0.284496736f);
    p = __builtin_fmaf(p, t, 0.254829592f);
    p = p * t;
    float e2 = __builtin_amdgcn_exp2f((ax * ax) * -LOG2EF); // exp(-x^2)
    float e  = __builtin_fmaf(-p, e2, 1.0f);
    return __builtin_copysignf(e, x);
}

__device__ __forceinline__ float exp_fast(float x) {
    return __builtin_amdgcn_exp2f(x * LOG2EF);
}

__device__ __forceinline__ v4f bs_price_one(float s0, float k, float t, float vt) {
    const float inv_sqrt2 = 0.7071067811865476f;

    float sv     = __builtin_amdgcn_sqrtf(vt * t);
    float inv_sv = __builtin_amdgcn_rcpf(sv);
    float ratio  = s0 * __builtin_amdgcn_rcpf(k);
    float lg     = __builtin_amdgcn_logf(ratio) * LN2F;      // ln(S0/K)
    float num    = __builtin_fmaf(__builtin_fmaf(0.5f, vt, BS_R - BS_Q), t, lg);
    float d1     = num * inv_sv;
    float d2     = d1 - sv;

    float n1 = __builtin_fmaf(0.5f, erf_fast(d1 * inv_sqrt2), 0.5f);
    float n2 = __builtin_fmaf(0.5f, erf_fast(d2 * inv_sqrt2), 0.5f);

    float disc_r = exp_fast(-BS_R * t);
    float disc_q = exp_fast(-BS_Q * t);

    float sdq = s0 * disc_q;
    float kdr = k * disc_r;
    float om1 = 1.0f - n1;
    float om2 = 1.0f - n2;

    float call = __builtin_fmaf(sdq, n1, -(kdr * n2));
    float put  = __builtin_fmaf(kdr, om2, -(sdq * om1));
    float dc   = disc_r * n2;
    float dp   = disc_r * om2;

    // Expired branch (T == 0): intrinsic / indicator payoffs (branchless).
    bool expired = (t == 0.0f);
    call = expired ? fmaxf(s0 - k, 0.0f) : call;
    put  = expired ? fmaxf(k - s0, 0.0f) : put;
    dc   = expired ? ((s0 > k) ? 1.0f : 0.0f) : dc;
    dp   = expired ? ((s0 <= k) ? 1.0f : 0.0f) : dp;

    v4f r;
    r.x = call; r.y = put; r.z = dc; r.w = dp;
    return r;
}

__global__ __launch_bounds__(256) void bs_all_kernel(
    const float* __restrict__ S0, const float* __restrict__ K,
    const float* __restrict__ T,  const float* __restrict__ VT,
    float* __restrict__ out, int n4, int n) {
    const int i = blockIdx.x * blockDim.x + threadIdx.x;

    if (i < n4) {
        const v4f* pS = (const v4f*)S0 + i;
        const v4f* pK = (const v4f*)K  + i;
        const v4f* pT = (const v4f*)T  + i;
        const v4f* pV = (const v4f*)VT + i;

        v4f s0, k, t, v;
        load4_b128_ht(pS, pK, pT, pV, s0, k, t, v);

        v4f o0 = bs_price_one(s0.x, k.x, t.x, v.x);
        v4f o1 = bs_price_one(s0.y, k.y, t.y, v.y);
        v4f o2 = bs_price_one(s0.z, k.z, t.z, v.z);
        v4f o3 = bs_price_one(s0.w, k.w, t.w, v.w);

        // Option j's 4 results are contiguous -> one b128 NT store each.
        v4f* po = (v4f*)out + 4 * (size_t)i;
        __builtin_nontemporal_store(o0, po + 0);
        __builtin_nontemporal_store(o1, po + 1);
        __builtin_nontemporal_store(o2, po + 2);
        __builtin_nontemporal_store(o3, po + 3);
    }

    // Scalar tail for n not divisible by 4 (at most 3 threads do work).
    int j = n4 * 4 + i;
    if (j < n) {
        v4f o = bs_price_one(S0[j], K[j], T[j], VT[j]);
        v4f* po = (v4f*)out + j;
        __builtin_nontemporal_store(o, po);
    }
}

extern "C" void kernel_launch(void* const* d_in, const int* in_sizes, int n_in,
                              void* d_out, int out_size, void* d_ws, size_t ws_size,
                              hipStream_t stream) {
    const float* S0 = (const float*)d_in[0];
    const float* K  = (const float*)d_in[1];
    const float* T  = (const float*)d_in[2];
    const float* VT = (const float*)d_in[3];
    float* out = (float*)d_out;

    const int n  = in_sizes[0];
    const int n4 = n / 4;

    const int block = 256;
    int grid = (n4 + block - 1) / block;
    if (grid < 1) grid = 1;
    bs_all_kernel<<<grid, block, 0, stream>>>(S0, K, T, VT, out, n4, n);
}